// QuantumLayer_62431644614770
// MI455X (gfx1250) — compile-verified
//
#include <hip/hip_runtime.h>
#include <math.h>

// ---------------------------------------------------------------------------
// 14-qubit batched state-vector simulator for MI455X (gfx1250).
//
//  * One workgroup (256 threads = 8 wave32) per batch element; the 16384-float
//    state lives entirely in LDS (320 KB/WGP makes this cheap).
//  * RY gates fused 4 wires at a time into 16x16 Kronecker matrices, applied
//    as dense 16x16x16 GEMMs with chained V_WMMA_F32_16X16X4_F32 (f32 exact).
//  * The adjacent-CNOT chain is the prefix-XOR basis permutation; its inverse
//    is the Gray code, folded into LDS addressing for free:
//      layer-2 RY passes address  n ^ (n>>1)
//      final probability pass     n ^ (n>>2)
//  * Z-expectation marginals fused into one squared-amplitude sweep with a
//    ds_add_f32 LDS reduction.
//  * Hardware transcendentals (v_sin_f32/v_cos_f32) for the gate angles —
//    inputs are O(1) radians, no libm argument-reduction path needed.
// ---------------------------------------------------------------------------

typedef __attribute__((ext_vector_type(2))) float v2f;
typedef __attribute__((ext_vector_type(8))) float v8f;

#define NQ     14
#define NSTATE 16384        // 2^14
#define BLK    256          // 8 waves of 32
#define NWAVES 8

// physical address of logical basis index n under the folded CNOT permutation
template <int MODE>
__device__ __forceinline__ int permA(int n) {
    if (MODE == 1) return n ^ (n >> 1);   // after one CNOT chain (gray)
    if (MODE == 2) return n ^ (n >> 2);   // after two CNOT chains (gray∘gray)
    return n;
}

// Apply 16x16 matrix U (row-major in LDS) along the 4-bit axis at bit offset F
// of the state index, in place.  D_tile = U x S_tile via 4 chained f32 WMMAs.
template <int MODE, int F>
__device__ __forceinline__ void apply_group(float* __restrict__ S,
                                            const float* __restrict__ U,
                                            int lane, int wave) {
    const int m    = lane & 15;          // A-matrix row / B,N column / D column
    const int hi   = lane >> 4;          // lane-half selects K (+2) / M (+8)
    const int kofs = hi ? 2 : 0;
    constexpr int lowm = (1 << F) - 1;

    // A operand = U, per documented 16x4 f32 layout:
    //   VGPR0: K = ks*4 + (hi?2:0), VGPR1: K = +1, rows M = lane%16
    v2f A[4];
#pragma unroll
    for (int ks = 0; ks < 4; ++ks) {
        int kb = ks * 4 + kofs;
        A[ks].x = U[m * 16 + kb];
        A[ks].y = U[m * 16 + kb + 1];
    }

    // 64 independent 16x16 state tiles, 8 per wave
#pragma unroll 2
    for (int t = wave; t < 64; t += NWAVES) {
        int o    = t * 16 + m;                         // "other" index (column)
        int base = ((o & ~lowm) << 4) | (o & lowm);    // n(o, k) = base + (k<<F)

        // B operand = state tile (K x N), K split across lane halves like A
        v2f Bv[4];
#pragma unroll
        for (int ks = 0; ks < 4; ++ks) {
            int k0 = ks * 4 + kofs;
            Bv[ks].x = S[permA<MODE>(base + (k0 << F))];
            Bv[ks].y = S[permA<MODE>(base + ((k0 + 1) << F))];
        }

        v8f d = {};
#pragma unroll
        for (int ks = 0; ks < 4; ++ks) {
            // D = A*B + C   (neg_a, A, neg_b, B, c_mod, C, reuse_a, reuse_b)
            d = __builtin_amdgcn_wmma_f32_16x16x4_f32(
                    false, A[ks], false, Bv[ks], (short)0, d, false, false);
        }

        // D layout: VGPR r -> row M = r + (hi?8:0), column N = lane%16
#pragma unroll
        for (int r = 0; r < 8; ++r) {
            int j = r + (hi ? 8 : 0);
            S[permA<MODE>(base + (j << F))] = d[r];
        }
    }
}

// Build the fused 16x16 matrix for one 4-wire group and apply it.
// Group G: field offset F in {10,6,2,0}; wires outside the group's real wire
// range get angle 0 (RY(0) = identity padding, used for the 2-wire tail group).
template <int MODE, int G>
__device__ __forceinline__ void do_pass(float* __restrict__ S,
                                        float* __restrict__ U,
                                        const float* __restrict__ src,
                                        int tid, int lane, int wave) {
    constexpr int F  = (G == 0) ? 10 : (G == 1) ? 6 : (G == 2) ? 2 : 0;
    constexpr int MW = (G == 0) ? 0 : (G == 1) ? 4 : (G == 2) ? 8 : 12;

    float cc[4], ss[4];
#pragma unroll
    for (int p = 0; p < 4; ++p) {
        constexpr_helper:;
        int   wi  = 13 - F - p;                       // wire on field bit p
        float ang = (wi >= MW) ? 0.5f * src[wi] : 0.0f;
        cc[p] = __cosf(ang);                          // v_cos_f32
        ss[p] = __sinf(ang);                          // v_sin_f32
    }

    // U[j][k] = prod_p R(theta_p)[j_p, k_p],  R = [[c,-s],[s,c]]
    {
        int   e = tid;                                // 256 threads, 256 entries
        int   j = e >> 4, k = e & 15;
        float v = 1.0f;
#pragma unroll
        for (int p = 0; p < 4; ++p) {
            int   jb = (j >> p) & 1, kb = (k >> p) & 1;
            float r  = (jb == kb) ? cc[p] : (jb ? ss[p] : -ss[p]);
            v *= r;
        }
        U[e] = v;
    }
    __syncthreads();                 // U visible to all waves
    apply_group<MODE, F>(S, U, lane, wave);
    __syncthreads();                 // state settled before next pass
}

template <int MODE>
__device__ __forceinline__ void do_layer(float* __restrict__ S,
                                         float* __restrict__ U,
                                         const float* __restrict__ src,
                                         int tid, int lane, int wave) {
    do_pass<MODE, 0>(S, U, src, tid, lane, wave);   // wires 0-3   (bits 13-10)
    do_pass<MODE, 1>(S, U, src, tid, lane, wave);   // wires 4-7   (bits 9-6)
    do_pass<MODE, 2>(S, U, src, tid, lane, wave);   // wires 8-11  (bits 5-2)
    do_pass<MODE, 3>(S, U, src, tid, lane, wave);   // wires 12-13 (bits 1-0, padded)
}

__global__ void __launch_bounds__(BLK)
qsim14_kernel(const float* __restrict__ x,       // [512,14]
              const float* __restrict__ w,       // [2,14]
              float* __restrict__ out) {         // [512,14]
    extern __shared__ float lds[];
    float* S   = lds;            // 16384 state amplitudes
    float* U   = lds + NSTATE;   // 256 fused gate matrix
    float* ACC = U + 256;        // 16 reduction slots

    const int tid  = threadIdx.x;
    const int lane = tid & 31;
    const int wave = tid >> 5;
    const int b    = blockIdx.x;

    // |0...0>
    for (int i = tid; i < NSTATE; i += BLK) S[i] = 0.0f;
    if (tid < 16) ACC[tid] = 0.0f;
    __syncthreads();
    if (tid == 0) S[0] = 1.0f;
    __syncthreads();

    // input encoding layer: RY(x[b,i])  (identity permutation)
    do_layer<0>(S, U, x + b * NQ, tid, lane, wave);

    // entangling layer 0: RY(weights[0,i]); CNOT chain folded into addressing
    do_layer<0>(S, U, w, tid, lane, wave);

    // entangling layer 1: operates through the gray permutation
    do_layer<1>(S, U, w + NQ, tid, lane, wave);

    // probs + all 14 Z marginals in one sweep (through gray∘gray permutation)
    float acc[NQ];
#pragma unroll
    for (int q = 0; q < NQ; ++q) acc[q] = 0.0f;
    for (int n = tid; n < NSTATE; n += BLK) {
        float v = S[n ^ (n >> 2)];
        float p = v * v;
#pragma unroll
        for (int q = 0; q < NQ; ++q)
            acc[q] += ((n >> (13 - q)) & 1) ? -p : p;
    }
#pragma unroll
    for (int q = 0; q < NQ; ++q) atomicAdd(&ACC[q], acc[q]);
    __syncthreads();
    if (tid < NQ) out[b * NQ + tid] = ACC[tid];
}

extern "C" void kernel_launch(void* const* d_in, const int* in_sizes, int n_in,
                              void* d_out, int out_size, void* d_ws, size_t ws_size,
                              hipStream_t stream) {
    const float* x = (const float*)d_in[0];   // [512,14] f32
    const float* w = (const float*)d_in[1];   // [2,14]   f32
    float* out = (float*)d_out;               // [512,14] f32

    const size_t shbytes = (size_t)(NSTATE + 256 + 16) * sizeof(float);
    qsim14_kernel<<<dim3(512), dim3(BLK), shbytes, stream>>>(x, w, out);
}